// Bow_Pooling_87625922773001
// MI455X (gfx1250) — compile-verified
//
#include <hip/hip_runtime.h>

typedef __attribute__((ext_vector_type(2))) float v2f;
typedef __attribute__((ext_vector_type(8))) float v8f;

constexpr int B_ = 8, C_ = 256, N_ = 4096, K_ = 2048;
constexpr int NCOL = 32;   // sim columns per workgroup
constexpr int SIMS = 34;   // simLDS row stride (pad: 34 -> conflict-free row & column access)
constexpr int INS  = 258;  // inT row stride (even for b64, conflict-free)
constexpr int LSEL = 1024; // int((N+1)*0.25)
constexpr int NTILES = N_ / NCOL; // 128

// Monotone float -> uint32 key: order(float) == order(uint)
__device__ __forceinline__ unsigned sortKey(float f) {
  unsigned x = __float_as_uint(f);
  return x ^ (unsigned)(((int)x >> 31) | 0x80000000);
}

__global__ __launch_bounds__(256) void bow_sim_kernel(
    const float* __restrict__ inputs, const float* __restrict__ dict,
    float* __restrict__ partial, float* __restrict__ out, int useAtomic)
{
  __shared__ float    simLDS[K_ * SIMS];      // 272 KB
  __shared__ float    inT[NCOL * INS];        // 32.25 KB
  __shared__ unsigned threshLDS[NCOL];

  const int nt = blockIdx.x;        // 0..127  (N tile)
  const int b  = blockIdx.y;        // 0..7    (batch)
  const int n0 = nt * NCOL;
  const int t  = threadIdx.x;
  const int lane = t & 31, wid = t >> 5;

  // ---- Stage input tile transposed: inT[n][c] = inputs[b][c][n0+n] ----
  {
    const int n = t & 31, c0 = t >> 5;           // lanes -> consecutive n: coalesced
    #pragma unroll 4
    for (int r = 0; r < 32; ++r) {
      const int c = r * 8 + c0;
      inT[n * INS + c] = inputs[((size_t)(b * C_ + c)) * N_ + (n0 + n)];
    }
  }
  __syncthreads();

  // ---- fp32 WMMA GEMM: sim[k][n] = dict[k,:] . inputs[b,:,n0+n] ----
  {
    const int m = lane & 15, half = lane >> 4;
    const float* aBase = dict + (size_t)m * C_ + 2 * half;
    const float* bBase = &inT[m * INS + 2 * half];
    for (int kt = wid; kt < K_ / 16; kt += 8) {   // 16 k-tiles per wave
      const int k0 = kt << 4;
      const float* aPtr = aBase + (size_t)k0 * C_;
      v8f acc0 = {}, acc1 = {};
      #pragma unroll 4
      for (int c = 0; c < C_; c += 4) {
        v2f af  = *(const v2f*)(aPtr + c);                   // global_load_b64 (dict)
        v2f bf0 = *(const v2f*)(bBase + c);                  // ds_load_b64
        v2f bf1 = *(const v2f*)(bBase + 16 * INS + c);       // ds_load_b64
        acc0 = __builtin_amdgcn_wmma_f32_16x16x4_f32(false, af, false, bf0,
                                                     (short)0, acc0, false, false);
        acc1 = __builtin_amdgcn_wmma_f32_16x16x4_f32(false, af, false, bf1,
                                                     (short)0, acc1, false, false);
      }
      #pragma unroll
      for (int r = 0; r < 8; ++r) {               // C/D layout: VGPR r = rows r, r+8
        const int row = k0 + r + 8 * half;
        simLDS[row * SIMS + m]      = acc0[r];
        simLDS[row * SIMS + 16 + m] = acc1[r];
      }
    }
  }
  __syncthreads();

  // ---- Per-column rank-1024 threshold: bitwise radix select over 2048 vals ----
  for (int cc = 0; cc < NCOL / 8; ++cc) {         // 4 columns per wave
    const int n = wid * 4 + cc;
    unsigned u[64];
    #pragma unroll
    for (int j = 0; j < 64; ++j)
      u[j] = sortKey(simLDS[(lane + 32 * j) * SIMS + n]);   // conflict-free (34*lane)
    unsigned T = 0;
    for (int bit = 31; bit >= 0; --bit) {
      const unsigned T2 = T | (1u << bit);
      int cnt = 0;
      #pragma unroll
      for (int j = 0; j < 64; ++j) cnt += (u[j] >= T2) ? 1 : 0;
      #pragma unroll
      for (int s = 1; s < 32; s <<= 1) cnt += __shfl_xor(cnt, s, 32);
      if (cnt >= LSEL) T = T2;    // invariant: count(u >= T) >= LSEL; final T = l-th largest
    }
    if (lane == 0) threshLDS[n] = T;
  }
  __syncthreads();

  // ---- Masked accumulation over the 32 columns, per dictionary row ----
  #pragma unroll
  for (int i = 0; i < 8; ++i) {
    const int k = t + 256 * i;                    // coalesced stores across threads
    float s = 0.f;
    #pragma unroll 8
    for (int n = 0; n < NCOL; ++n) {
      const float v = simLDS[k * SIMS + n];
      if (sortKey(v) >= threshLDS[n]) s += v;     // same >= semantics as reference mask
    }
    if (useAtomic) atomicAdd(&out[b * K_ + k], s);
    else           partial[((size_t)(b * NTILES + nt)) * K_ + k] = s;
  }
}

// Deterministic fixed-order reduction of partials: out[b][k] = sum_nt partial[b][nt][k]
__global__ __launch_bounds__(256) void bow_reduce_kernel(
    const float* __restrict__ partial, float* __restrict__ out)
{
  const int idx = blockIdx.x * blockDim.x + threadIdx.x;   // 0..16383
  const int b = idx >> 11, k = idx & (K_ - 1);
  const float* p = partial + (size_t)b * NTILES * K_ + k;
  float s = 0.f;
  for (int ntile = 0; ntile < NTILES; ++ntile) s += p[(size_t)ntile * K_];
  out[idx] = s;
}

__global__ __launch_bounds__(256) void bow_zero_kernel(float* __restrict__ out) {
  out[blockIdx.x * blockDim.x + threadIdx.x] = 0.f;
}

extern "C" void kernel_launch(void* const* d_in, const int* in_sizes, int n_in,
                              void* d_out, int out_size, void* d_ws, size_t ws_size,
                              hipStream_t stream) {
  const float* inputs = (const float*)d_in[0];   // [8,256,4096] f32
  const float* dict   = (const float*)d_in[1];   // [2048,256]  f32
  float* out = (float*)d_out;                    // [8,2048]    f32
  float* partial = (float*)d_ws;

  const size_t need = (size_t)B_ * NTILES * K_ * sizeof(float); // 8 MB
  dim3 grid(NTILES, B_), block(256);
  if (ws_size >= need) {
    bow_sim_kernel<<<grid, block, 0, stream>>>(inputs, dict, partial, out, 0);
    bow_reduce_kernel<<<(B_ * K_) / 256, 256, 0, stream>>>(partial, out);
  } else {
    bow_zero_kernel<<<(B_ * K_) / 256, 256, 0, stream>>>(out);
    bow_sim_kernel<<<grid, block, 0, stream>>>(inputs, dict, nullptr, out, 1);
  }
}